// FixStepModel_62775241998343
// MI455X (gfx1250) — compile-verified
//
#include <hip/hip_runtime.h>
#include <math.h>

// ---------------- model constants ----------------
#define Bq 1024
#define NUME 20000
#define NUMR 8
#define NET 17           // 2*NUMR+1
#define DIMT 28
#define DIME 100
#define DIMD 200         // DIM
#define HEADS 4
#define OHh 50
#define N0c 160000
#define N1c 40000
#define Rc 8192          // 2*B*H
#define E0c 60000
#define E1c 16000
#define KCOPY 500000
#define DIMR 100
#define HATT 4
#define OA 50
#define DIN0 128         // DIME+DIMT
#define DIN1 228         // DIM+DIMT
#define BM  ((size_t)Bq * NUME)   // one output matrix, 20.48M floats

typedef float v8f __attribute__((ext_vector_type(8)));
typedef float v2f __attribute__((ext_vector_type(2)));

static inline int cdiv(long a, long b) { return (int)((a + b - 1) / b); }

// ---------- monotonic float<->uint mapping for atomic max ----------
__device__ __forceinline__ unsigned fmapu(float f) {
  unsigned u = __float_as_uint(f);
  return (u & 0x80000000u) ? ~u : (u | 0x80000000u);
}
__device__ __forceinline__ float funmap2(unsigned u) {
  return (u & 0x80000000u) ? __uint_as_float(u & 0x7FFFFFFFu)
                           : __uint_as_float(~u);
}

__device__ __forceinline__ float lrelu02(float x) { return x > 0.f ? x : 0.2f * x; }

// ---------------- generic fill ----------------
__global__ void k_fill_f32(float* p, float v, size_t n) {
  size_t i = (size_t)blockIdx.x * blockDim.x + threadIdx.x;
  size_t st = (size_t)gridDim.x * blockDim.x;
  for (; i < n; i += st) p[i] = v;
}

// ---------------- stage 0: x0 = [emb_table[nid%NUME], time_emb] ----------------
__device__ __forceinline__ float temb_val(int nid, int ts, const float* tf, const float* tp, int j) {
  float t = (float)(ts - nid / NUME);
  return cosf(t * tf[j] + tp[j]);
}

__global__ void k_x0(const float* __restrict__ emb, const int* __restrict__ nid,
                     const int* __restrict__ tsp, const float* __restrict__ tf,
                     const float* __restrict__ tp, float* __restrict__ x0) {
  size_t idx = (size_t)blockIdx.x * blockDim.x + threadIdx.x;
  if (idx >= (size_t)N0c * DIN0) return;
  int n = (int)(idx / DIN0), c = (int)(idx % DIN0);
  int nd = nid[n];
  if (c < DIME) x0[idx] = emb[(size_t)(nd % NUME) * DIME + c];
  else          x0[idx] = temb_val(nd, tsp[0], tf, tp, c - DIME);
}

// ---------------- precontract attention weights ----------------
// wal[r,i,h] = sum_o Wg[r,i,h,o]*al[r,h,o] ; war analogous with ar.
__global__ void k_redw(const float* __restrict__ Wg, const float* __restrict__ al,
                       const float* __restrict__ ar, float* __restrict__ wal,
                       float* __restrict__ war, int DIN) {
  int idx = blockIdx.x * blockDim.x + threadIdx.x;
  if (idx >= NET * DIN) return;
  int r = idx / DIN, i = idx % DIN;
  for (int h = 0; h < HEADS; ++h) {
    const float* wrow = Wg + ((size_t)(r * DIN + i)) * (HEADS * OHh) + h * OHh;
    const float* alr = al + (size_t)r * (HEADS * OHh) + h * OHh;
    const float* arr = ar + (size_t)r * (HEADS * OHh) + h * OHh;
    float sl = 0.f, sr = 0.f;
    for (int o = 0; o < OHh; ++o) { sl += wrow[o] * alr[o]; sr += wrow[o] * arr[o]; }
    wal[(size_t)idx * HEADS + h] = sl;
    war[(size_t)idx * HEADS + h] = sr;
  }
}

__global__ void k_bgs(const float* __restrict__ bg, float* __restrict__ bgs) {
  int c = blockIdx.x * blockDim.x + threadIdx.x;
  if (c >= DIMD) return;
  float s = 0.f;
  for (int r = 0; r < NET; ++r) s += bg[(size_t)r * DIMD + c];
  bgs[c] = s;
}

// ---------------- per-node right-attention term ----------------
__global__ void k_er(const float* __restrict__ x, const float* __restrict__ war,
                     float* __restrict__ er, int DIN, int nd) {
  int idx = blockIdx.x * blockDim.x + threadIdx.x;
  if (idx >= NET * nd) return;
  int r = idx / nd, n = idx % nd;
  const float* xr = x + (size_t)n * DIN;
  const float* wr = war + (size_t)r * DIN * HEADS;
  float a0 = 0.f, a1 = 0.f, a2 = 0.f, a3 = 0.f;
  for (int i = 0; i < DIN; ++i) {
    float xv = xr[i];
    a0 += xv * wr[i * 4 + 0]; a1 += xv * wr[i * 4 + 1];
    a2 += xv * wr[i * 4 + 2]; a3 += xv * wr[i * 4 + 3];
  }
  float* o = er + (size_t)idx * 4;
  o[0] = a0; o[1] = a1; o[2] = a2; o[3] = a3;
}

// ---------------- edge logits + segment max ----------------
__global__ void k_elog(const float* __restrict__ x, const float* __restrict__ wal,
                       const float* __restrict__ er, const int* __restrict__ src,
                       const int* __restrict__ dst, float* __restrict__ ebuf,
                       unsigned* __restrict__ mmax, int E, int nd, int DIN) {
  int idx = blockIdx.x * blockDim.x + threadIdx.x;
  if (idx >= NET * E) return;
  int r = idx / E, e = idx % E;
  int s = src[(size_t)r * E + e];
  int d = dst[(size_t)r * E + e];
  const float* xr = x + (size_t)s * DIN;
  const float* wr = wal + (size_t)r * DIN * HEADS;
  float a0 = 0.f, a1 = 0.f, a2 = 0.f, a3 = 0.f;
  for (int i = 0; i < DIN; ++i) {
    float xv = xr[i];
    a0 += xv * wr[i * 4 + 0]; a1 += xv * wr[i * 4 + 1];
    a2 += xv * wr[i * 4 + 2]; a3 += xv * wr[i * 4 + 3];
  }
  float acc[4] = {a0, a1, a2, a3};
  const float* ern = er + ((size_t)r * nd + d) * 4;
  float* eo = ebuf + (size_t)idx * 4;
  unsigned* mo = mmax + ((size_t)r * nd + d) * 4;
  for (int h = 0; h < 4; ++h) {
    float v = lrelu02(acc[h] + ern[h]);
    eo[h] = v;
    atomicMax(&mo[h], fmapu(v));
  }
}

// ---------------- exp / segment sum ----------------
__global__ void k_ealpha(float* __restrict__ ebuf, const unsigned* __restrict__ mmax,
                         float* __restrict__ z, const int* __restrict__ dst,
                         int E, int nd) {
  int idx = blockIdx.x * blockDim.x + threadIdx.x;
  if (idx >= NET * E) return;
  int r = idx / E, e = idx % E;
  int d = dst[(size_t)r * E + e];
  float* eo = ebuf + (size_t)idx * 4;
  const unsigned* mo = mmax + ((size_t)r * nd + d) * 4;
  float* zo = z + ((size_t)r * nd + d) * 4;
  for (int h = 0; h < 4; ++h) {
    float a = expf(eo[h] - funmap2(mo[h]));
    eo[h] = a;
    atomicAdd(&zo[h], a);
  }
}

// ---------------- normalize: ebuf <- a / z[dst]  (hoists div out of WMMA loop) ----------------
__global__ void k_adiv(float* __restrict__ ebuf, const float* __restrict__ z,
                       const int* __restrict__ dst, int E, int nd) {
  int idx = blockIdx.x * blockDim.x + threadIdx.x;
  if (idx >= NET * E) return;
  int r = idx / E, e = idx % E;
  int d = dst[(size_t)r * E + e];
  float* eo = ebuf + (size_t)idx * 4;
  const float* zo = z + ((size_t)r * nd + d) * 4;
  for (int h = 0; h < 4; ++h) eo[h] = eo[h] / zo[h];
}

// ---------------- WMMA gathered-edge projection + weighted scatter-add ----------------
// One wave per 16 edges of one relation. A = gathered x[src] rows (cached in regs),
// B = Wg[r] (DIN x 200), C tiles 16x16 over the 200 output cols; each C element is
// scaled by alpha (pre-normalized) and atomically accumulated into msg[dst].
// Out-of-range tile columns load from a clamped column (results discarded), so the
// inner WMMA loop has no divergence.
template <int DIN>
__global__ __launch_bounds__(32)
void k_msg(const float* __restrict__ x, const float* __restrict__ Wg,
           const float* __restrict__ alpha, const int* __restrict__ src,
           const int* __restrict__ dst, float* __restrict__ msg, int E) {
  constexpr int KS = DIN / 4;
  int lane = threadIdx.x;
  int r = blockIdx.y;
  int e0 = blockIdx.x * 16;
  int mr = lane & 15;
  int khalf = (lane >> 4) * 2;

  int srow = src[(size_t)r * E + e0 + mr];
  const float* xr = x + (size_t)srow * DIN;
  v2f ka[KS];
#pragma unroll
  for (int t = 0; t < KS; ++t) {
    ka[t][0] = xr[4 * t + khalf + 0];
    ka[t][1] = xr[4 * t + khalf + 1];
  }
  int dct[8];
#pragma unroll
  for (int v = 0; v < 8; ++v)
    dct[v] = dst[(size_t)r * E + e0 + v + 8 * (lane >> 4)];

  const float* Wr = Wg + (size_t)r * DIN * DIMD;
  for (int n0 = 0; n0 < 208; n0 += 16) {   // 13 tiles cover 200 cols
    int col = n0 + mr;
    bool okc = col < DIMD;
    int colc = okc ? col : (DIMD - 1);     // clamp: straight-line loads, no EXEC dance
    v8f acc = {0.f, 0.f, 0.f, 0.f, 0.f, 0.f, 0.f, 0.f};
#pragma unroll
    for (int t = 0; t < KS; ++t) {
      v2f bb;
      bb[0] = Wr[(size_t)(4 * t + khalf + 0) * DIMD + colc];
      bb[1] = Wr[(size_t)(4 * t + khalf + 1) * DIMD + colc];
      acc = __builtin_amdgcn_wmma_f32_16x16x4_f32(false, ka[t], false, bb,
                                                  (short)0, acc, false, false);
    }
    if (okc) {
      int h = col / OHh;
#pragma unroll
      for (int v = 0; v < 8; ++v) {
        int e = e0 + v + 8 * (lane >> 4);
        float a = alpha[((size_t)r * E + e) * 4 + h];
        atomicAdd(&msg[(size_t)dct[v] * DIMD + col], a * acc[v]);
      }
    }
  }
}

// ---------------- layer epilogues ----------------
__global__ void k_x1(const float* __restrict__ msg0, const float* __restrict__ bgs0,
                     const int* __restrict__ nid, const int* __restrict__ tsp,
                     const float* __restrict__ tf, const float* __restrict__ tp,
                     float* __restrict__ x1) {
  size_t idx = (size_t)blockIdx.x * blockDim.x + threadIdx.x;
  if (idx >= (size_t)N1c * DIN1) return;
  int n = (int)(idx / DIN1), c = (int)(idx % DIN1);
  if (c < DIMD) x1[idx] = fmaxf((msg0[(size_t)n * DIMD + c] + bgs0[c]) / (float)NET, 0.f);
  else          x1[idx] = temb_val(nid[n], tsp[0], tf, tp, c - DIMD);
}

__global__ void k_h2(const float* __restrict__ msg1, const float* __restrict__ bgs1,
                     float* __restrict__ h2) {
  size_t idx = (size_t)blockIdx.x * blockDim.x + threadIdx.x;
  if (idx >= (size_t)Rc * DIMD) return;
  int c = (int)(idx % DIMD);
  h2[idx] = fmaxf((msg1[idx] + bgs1[c]) / (float)NET, 0.f);
}

__global__ void k_hid(const float* __restrict__ h2, const int* __restrict__ root,
                      float* __restrict__ hid) {
  size_t idx = (size_t)blockIdx.x * blockDim.x + threadIdx.x;
  if (idx >= (size_t)2 * Bq * (HEADS * DIMD)) return;
  int i = (int)(idx / 800), j = (int)(idx % 800);
  int rid = root[i * 4 + j / DIMD];
  hid[idx] = h2[(size_t)rid * DIMD + (j % DIMD)];
}

// ---------------- layernorm (one block per row) ----------------
__global__ __launch_bounds__(256)
void k_ln(const float* __restrict__ in, const float* __restrict__ w,
          const float* __restrict__ b, float* __restrict__ out, int width) {
  __shared__ float sm[256], sv[256];
  int row = blockIdx.x, tid = threadIdx.x;
  const float* ir = in + (size_t)row * width;
  float s = 0.f, s2 = 0.f;
  for (int i = tid; i < width; i += 256) { float v = ir[i]; s += v; s2 += v * v; }
  sm[tid] = s; sv[tid] = s2;
  __syncthreads();
  for (int off = 128; off > 0; off >>= 1) {
    if (tid < off) { sm[tid] += sm[tid + off]; sv[tid] += sv[tid + off]; }
    __syncthreads();
  }
  float mean = sm[0] / width;
  float var = sv[0] / width - mean * mean;
  float inv = rsqrtf(var + 1e-5f);
  float* orow = out + (size_t)row * width;
  for (int i = tid; i < width; i += 256)
    orow[i] = (ir[i] - mean) * inv * w[i] + b[i];
}

// ---------------- tiny 4x4 self-attention, one block per (b,h) ----------------
__global__ __launch_bounds__(64)
void k_att(const float* __restrict__ xln, const float* __restrict__ Wq,
           const float* __restrict__ bq, const float* __restrict__ Wv,
           const float* __restrict__ bv, float* __restrict__ x2) {
  int bh = blockIdx.x;
  int b = bh >> 2, h = bh & 3;
  __shared__ float q[200], v[200], lg[16], att[16];
  int tid = threadIdx.x;
  const float* xb = xln + (size_t)b * 800;
  const float* Wqh = Wq + (size_t)h * DIMD * OA;
  const float* Wvh = Wv + (size_t)h * DIMD * OA;
  for (int idx = tid; idx < 200; idx += 64) {
    int s = idx / OA, o = idx % OA;
    float aq = bq[h * OA + o], av = bv[h * OA + o];
    const float* xs = xb + s * DIMD;
    for (int e = 0; e < DIMD; ++e) {
      float xv = xs[e];
      aq += xv * Wqh[(size_t)e * OA + o];
      av += xv * Wvh[(size_t)e * OA + o];
    }
    q[idx] = aq; v[idx] = av;
  }
  __syncthreads();
  if (tid < 16) {
    int s = tid >> 2, t = tid & 3;
    float acc = 0.f;
    for (int o = 0; o < OA; ++o) acc += q[s * OA + o] * q[t * OA + o];
    lg[s * 4 + t] = acc;
  }
  __syncthreads();
  if (tid < 4) {  // softmax over s for fixed t (axis=2 in reference)
    int t = tid;
    float mx = -1e30f;
    for (int s = 0; s < 4; ++s) mx = fmaxf(mx, lg[s * 4 + t]);
    float ex[4], sum = 0.f;
    for (int s = 0; s < 4; ++s) { ex[s] = expf(lg[s * 4 + t] - mx); sum += ex[s]; }
    for (int s = 0; s < 4; ++s) att[s * 4 + t] = ex[s] / sum;
  }
  __syncthreads();
  for (int idx = tid; idx < 200; idx += 64) {
    int s = idx / OA, o = idx % OA;
    float acc = 0.f;
    for (int t = 0; t < 4; ++t) acc += att[s * 4 + t] * v[t * OA + o];
    x2[(size_t)b * 800 + s * DIMD + h * OA + o] = fmaxf(acc, 0.f);  // fused relu
  }
}

// ---------------- WMMA GEMM: xfin = relu(xln1 @ Wd1 + bd1)  (2048x800x200) ----------------
__global__ __launch_bounds__(32)
void k_gemm_wd1(const float* __restrict__ A, const float* __restrict__ W,
                const float* __restrict__ bias, float* __restrict__ C) {
  int lane = threadIdx.x;
  int n0 = blockIdx.x * 16;       // 13 tiles over 200
  int m0 = blockIdx.y * 16;       // 128 tiles over 2048
  int mr = lane & 15;
  int khalf = (lane >> 4) * 2;
  int col = n0 + mr;
  bool okc = col < DIMD;
  int colc = okc ? col : (DIMD - 1);   // clamp for straight-line loads
  const float* Ar = A + (size_t)(m0 + mr) * 800;
  v8f acc = {0.f, 0.f, 0.f, 0.f, 0.f, 0.f, 0.f, 0.f};
  for (int t = 0; t < 200; ++t) {   // K = 800
    v2f a, b;
    a[0] = Ar[4 * t + khalf + 0];
    a[1] = Ar[4 * t + khalf + 1];
    b[0] = W[(size_t)(4 * t + khalf + 0) * DIMD + colc];
    b[1] = W[(size_t)(4 * t + khalf + 1) * DIMD + colc];
    acc = __builtin_amdgcn_wmma_f32_16x16x4_f32(false, a, false, b,
                                                (short)0, acc, false, false);
  }
  if (okc) {
    float bb = bias[col];
#pragma unroll
    for (int v = 0; v < 8; ++v)
      C[(size_t)(m0 + v + 8 * (lane >> 4)) * DIMD + col] = fmaxf(acc[v] + bb, 0.f);
  }
}

// ---------------- build concat inputs for prediction heads ----------------
__global__ void k_catA(const float* __restrict__ xf, int rowoff,
                       const float* __restrict__ relemb, const int* __restrict__ rel,
                       float* __restrict__ A) {
  size_t idx = (size_t)blockIdx.x * blockDim.x + threadIdx.x;
  if (idx >= (size_t)Bq * 300) return;
  int i = (int)(idx / 300), j = (int)(idx % 300);
  A[idx] = (j < DIMD) ? xf[(size_t)(rowoff + i) * DIMD + j]
                      : relemb[(size_t)rel[i] * DIMR + (j - DIMD)];
}

// ---------------- WMMA GEMM: pred = A(1024x300) @ W(300x20000) + bias ----------------
// A fragments register-cached; each wave computes TWO adjacent 16x16 N-tiles to
// reuse A (halves A traffic + wave count). Prefetch the B stream 8 K-steps ahead.
__global__ __launch_bounds__(32)
void k_gemm_pred(const float* __restrict__ A, const float* __restrict__ W,
                 const float* __restrict__ bias, float* __restrict__ C) {
  int lane = threadIdx.x;
  int n0 = blockIdx.x * 32;       // 625 double-tiles over 20000
  int m0 = blockIdx.y * 16;       // 64 tiles over 1024
  int mr = lane & 15;
  int khalf = (lane >> 4) * 2;
  const float* Ar = A + (size_t)(m0 + mr) * 300;
  v2f ka[75];
#pragma unroll
  for (int t = 0; t < 75; ++t) {
    ka[t][0] = Ar[4 * t + khalf + 0];
    ka[t][1] = Ar[4 * t + khalf + 1];
  }
  int c0 = n0 + mr;
  int c1 = n0 + 16 + mr;
  v8f acc0 = {0.f, 0.f, 0.f, 0.f, 0.f, 0.f, 0.f, 0.f};
  v8f acc1 = {0.f, 0.f, 0.f, 0.f, 0.f, 0.f, 0.f, 0.f};
#pragma unroll
  for (int t = 0; t < 75; ++t) {
    const float* Wr0 = W + (size_t)(4 * t + khalf + 0) * NUME;
    const float* Wr1 = W + (size_t)(4 * t + khalf + 1) * NUME;
    if (t + 8 < 75) {
      __builtin_prefetch(W + (size_t)(4 * (t + 8) + khalf) * NUME + c0, 0, 0);
    }
    v2f b0, b1;
    b0[0] = Wr0[c0]; b0[1] = Wr1[c0];
    b1[0] = Wr0[c1]; b1[1] = Wr1[c1];
    acc0 = __builtin_amdgcn_wmma_f32_16x16x4_f32(false, ka[t], false, b0,
                                                 (short)0, acc0, false, false);
    acc1 = __builtin_amdgcn_wmma_f32_16x16x4_f32(false, ka[t], false, b1,
                                                 (short)0, acc1, false, false);
  }
  float bb0 = bias[c0];
  float bb1 = bias[c1];
#pragma unroll
  for (int v = 0; v < 8; ++v) {
    size_t rowo = (size_t)(m0 + v + 8 * (lane >> 4)) * NUME;
    C[rowo + c0] = acc0[v] + bb0;
    C[rowo + c1] = acc1[v] + bb1;
  }
}

// ---------------- masked copy-logit gather (L2-resident cW matrices) ----------------
__global__ __launch_bounds__(256)
void k_scatter(const int* __restrict__ rows, const int* __restrict__ cols,
               const float* __restrict__ hid, const int* __restrict__ rel,
               const float* __restrict__ csrel, const float* __restrict__ corel,
               const float* __restrict__ cWsub, const float* __restrict__ cbsub,
               const float* __restrict__ cWobj, const float* __restrict__ cbobj,
               float* __restrict__ outm /* = csub base; cobj follows */) {
  int wid = (blockIdx.x * 256 + threadIdx.x) >> 5;
  int lane = threadIdx.x & 31;
  if (wid >= KCOPY) return;
  int row = rows[wid], col = cols[wid];
  const float *W, *bias, *re, *chrow;
  float* dstp;
  if (row < Bq) {            // raw_sub row: uses ch[B+row], corel, cWsub -> csub
    chrow = hid + (size_t)(Bq + row) * 800 + 600;
    re = corel + (size_t)rel[row] * DIMR;
    W = cWsub; bias = cbsub;
    dstp = outm + (size_t)row * NUME + col;
  } else {                   // raw_obj row: uses ch[row-B], csrel, cWobj -> cobj
    int r2 = row - Bq;
    chrow = hid + (size_t)r2 * 800 + 600;
    re = csrel + (size_t)rel[r2] * DIMR;
    W = cWobj; bias = cbobj;
    dstp = outm + BM + (size_t)r2 * NUME + col;
  }
  float s = 0.f;
  for (int j = lane; j < 300; j += 32) {
    float av = (j < DIMD) ? chrow[j] : re[j - DIMD];
    s += av * W[(size_t)j * NUME + col];
  }
  for (int off = 16; off; off >>= 1) s += __shfl_xor(s, off, 32);
  if (lane == 0) *dstp = s + bias[col];
}

// ================================================================
extern "C" void kernel_launch(void* const* d_in, const int* in_sizes, int n_in,
                              void* d_out, int out_size, void* d_ws, size_t ws_size,
                              hipStream_t stream) {
  (void)in_sizes; (void)n_in; (void)out_size; (void)ws_size;
  // inputs (setup_inputs dict order)
  const int*   rel       = (const int*)d_in[2];
  const int*   tsp       = (const int*)d_in[3];
  const int*   nid0      = (const int*)d_in[4];
  const int*   root_idx  = (const int*)d_in[5];
  const int*   src0      = (const int*)d_in[6];
  const int*   dst0      = (const int*)d_in[7];
  const int*   src1      = (const int*)d_in[8];
  const int*   dst1      = (const int*)d_in[9];
  const int*   copy_rows = (const int*)d_in[10];
  const int*   copy_cols = (const int*)d_in[11];
  const float* emb       = (const float*)d_in[12];
  const float* tf        = (const float*)d_in[13];
  const float* tp        = (const float*)d_in[14];
  const float* Wg0       = (const float*)d_in[15];
  const float* al0       = (const float*)d_in[16];
  const float* ar0       = (const float*)d_in[17];
  const float* bg0       = (const float*)d_in[18];
  const float* Wg1       = (const float*)d_in[19];
  const float* al1       = (const float*)d_in[20];
  const float* ar1       = (const float*)d_in[21];
  const float* bg1       = (const float*)d_in[22];
  const float* ln0w      = (const float*)d_in[23];
  const float* ln0b      = (const float*)d_in[24];
  const float* Wq        = (const float*)d_in[25];
  const float* bq        = (const float*)d_in[26];
  const float* Wv        = (const float*)d_in[27];
  const float* bv        = (const float*)d_in[28];
  const float* ln1w      = (const float*)d_in[29];
  const float* ln1b      = (const float*)d_in[30];
  const float* Wd1       = (const float*)d_in[31];
  const float* bd1       = (const float*)d_in[32];
  const float* srel      = (const float*)d_in[33];
  const float* orel      = (const float*)d_in[34];
  const float* Wsub      = (const float*)d_in[35];
  const float* bsub      = (const float*)d_in[36];
  const float* Wobj      = (const float*)d_in[37];
  const float* bobj      = (const float*)d_in[38];
  const float* csrel     = (const float*)d_in[39];
  const float* corel     = (const float*)d_in[40];
  const float* cWsub     = (const float*)d_in[41];
  const float* cbsub     = (const float*)d_in[42];
  const float* cWobj     = (const float*)d_in[43];
  const float* cbobj     = (const float*)d_in[44];

  float* out = (float*)d_out;

  // workspace carve-out (256B aligned)
  char* base = (char*)d_ws;
  size_t off = 0;
  auto take = [&](size_t bytes) -> char* {
    char* p = base + off;
    off = (off + bytes + 255) & ~(size_t)255;
    return p;
  };
  float*    x0   = (float*)take((size_t)N0c * DIN0 * 4);
  float*    wal0 = (float*)take((size_t)NET * DIN0 * 4 * 4);
  float*    war0 = (float*)take((size_t)NET * DIN0 * 4 * 4);
  float*    wal1 = (float*)take((size_t)NET * DIN1 * 4 * 4);
  float*    war1 = (float*)take((size_t)NET * DIN1 * 4 * 4);
  float*    bgs0 = (float*)take(DIMD * 4);
  float*    bgs1 = (float*)take(DIMD * 4);
  float*    er0  = (float*)take((size_t)NET * N1c * 4 * 4);
  float*    eb0  = (float*)take((size_t)NET * E0c * 4 * 4);
  unsigned* m0   = (unsigned*)take((size_t)NET * N1c * 4 * 4);
  float*    z0   = (float*)take((size_t)NET * N1c * 4 * 4);
  float*    msg0 = (float*)take((size_t)N1c * DIMD * 4);
  float*    x1   = (float*)take((size_t)N1c * DIN1 * 4);
  float*    er1  = (float*)take((size_t)NET * Rc * 4 * 4);
  float*    eb1  = (float*)take((size_t)NET * E1c * 4 * 4);
  unsigned* m1   = (unsigned*)take((size_t)NET * Rc * 4 * 4);
  float*    z1   = (float*)take((size_t)NET * Rc * 4 * 4);
  float*    msg1 = (float*)take((size_t)Rc * DIMD * 4);
  float*    h2   = (float*)take((size_t)Rc * DIMD * 4);
  float*    hid  = (float*)take((size_t)2 * Bq * 800 * 4);
  float*    xln  = (float*)take((size_t)2 * Bq * 800 * 4);
  float*    x2   = (float*)take((size_t)2 * Bq * 800 * 4);
  float*    xln1 = (float*)take((size_t)2 * Bq * 800 * 4);
  float*    xfin = (float*)take((size_t)2 * Bq * DIMD * 4);
  float*    Asub = (float*)take((size_t)Bq * 300 * 4);
  float*    Aobj = (float*)take((size_t)Bq * 300 * 4);

  const int T = 256;

  // stage 0: node features
  k_x0<<<cdiv((long)N0c * DIN0, T), T, 0, stream>>>(emb, nid0, tsp, tf, tp, x0);

  // precontract attention weight vectors + bias sums
  k_redw<<<cdiv(NET * DIN0, T), T, 0, stream>>>(Wg0, al0, ar0, wal0, war0, DIN0);
  k_redw<<<cdiv(NET * DIN1, T), T, 0, stream>>>(Wg1, al1, ar1, wal1, war1, DIN1);
  k_bgs<<<1, DIMD, 0, stream>>>(bg0, bgs0);
  k_bgs<<<1, DIMD, 0, stream>>>(bg1, bgs1);

  // ---- GAT layer 0 ----
  k_fill_f32<<<2048, T, 0, stream>>>((float*)m0, 0.0f, (size_t)NET * N1c * 4);
  k_fill_f32<<<2048, T, 0, stream>>>(z0, 0.0f, (size_t)NET * N1c * 4);
  k_fill_f32<<<2048, T, 0, stream>>>(msg0, 0.0f, (size_t)N1c * DIMD);
  k_er<<<cdiv((long)NET * N1c, T), T, 0, stream>>>(x0, war0, er0, DIN0, N1c);
  k_elog<<<cdiv((long)NET * E0c, T), T, 0, stream>>>(x0, wal0, er0, src0, dst0, eb0, m0, E0c, N1c, DIN0);
  k_ealpha<<<cdiv((long)NET * E0c, T), T, 0, stream>>>(eb0, m0, z0, dst0, E0c, N1c);
  k_adiv<<<cdiv((long)NET * E0c, T), T, 0, stream>>>(eb0, z0, dst0, E0c, N1c);
  k_msg<DIN0><<<dim3(E0c / 16, NET), 32, 0, stream>>>(x0, Wg0, eb0, src0, dst0, msg0, E0c);
  k_x1<<<cdiv((long)N1c * DIN1, T), T, 0, stream>>>(msg0, bgs0, nid0, tsp, tf, tp, x1);

  // ---- GAT layer 1 ----
  k_fill_f32<<<2048, T, 0, stream>>>((float*)m1, 0.0f, (size_t)NET * Rc * 4);
  k_fill_f32<<<2048, T, 0, stream>>>(z1, 0.0f, (size_t)NET * Rc * 4);
  k_fill_f32<<<2048, T, 0, stream>>>(msg1, 0.0f, (size_t)Rc * DIMD);
  k_er<<<cdiv((long)NET * Rc, T), T, 0, stream>>>(x1, war1, er1, DIN1, Rc);
  k_elog<<<cdiv((long)NET * E1c, T), T, 0, stream>>>(x1, wal1, er1, src1, dst1, eb1, m1, E1c, Rc, DIN1);
  k_ealpha<<<cdiv((long)NET * E1c, T), T, 0, stream>>>(eb1, m1, z1, dst1, E1c, Rc);
  k_adiv<<<cdiv((long)NET * E1c, T), T, 0, stream>>>(eb1, z1, dst1, E1c, Rc);
  k_msg<DIN1><<<dim3(E1c / 16, NET), 32, 0, stream>>>(x1, Wg1, eb1, src1, dst1, msg1, E1c);
  k_h2<<<cdiv((long)Rc * DIMD, T), T, 0, stream>>>(msg1, bgs1, h2);

  // gather roots
  k_hid<<<cdiv((long)2 * Bq * 800, T), T, 0, stream>>>(h2, root_idx, hid);

  // ---- masked copy-logits: fill -100 then gather the K live entries ----
  k_fill_f32<<<4096, T, 0, stream>>>(out + 2 * BM, -100.0f, 2 * BM);
  k_scatter<<<cdiv((long)KCOPY * 32, T), T, 0, stream>>>(copy_rows, copy_cols, hid, rel,
                                                         csrel, corel, cWsub, cbsub,
                                                         cWobj, cbobj, out + 2 * BM);

  // ---- attention block ----
  k_ln<<<2 * Bq, 256, 0, stream>>>(hid, ln0w, ln0b, xln, 800);
  k_att<<<2 * Bq * HATT, 64, 0, stream>>>(xln, Wq, bq, Wv, bv, x2);
  k_ln<<<2 * Bq, 256, 0, stream>>>(x2, ln1w, ln1b, xln1, 800);
  k_gemm_wd1<<<dim3(13, 2 * Bq / 16), 32, 0, stream>>>(xln1, Wd1, bd1, xfin);

  // ---- prediction heads ----
  k_catA<<<cdiv((long)Bq * 300, T), T, 0, stream>>>(xfin, Bq, orel, rel, Asub);
  k_catA<<<cdiv((long)Bq * 300, T), T, 0, stream>>>(xfin, 0, srel, rel, Aobj);
  k_gemm_pred<<<dim3(NUME / 32, Bq / 16), 32, 0, stream>>>(Asub, Wsub, bsub, out);
  k_gemm_pred<<<dim3(NUME / 32, Bq / 16), 32, 0, stream>>>(Aobj, Wobj, bobj, out + BM);
}